// SpatialTransformer_23433341567034
// MI455X (gfx1250) — compile-verified
//
#include <hip/hip_runtime.h>
#include <stdint.h>

typedef float v2f __attribute__((ext_vector_type(2)));

#define BATCH 2
#define DD 160
#define HH 192
#define WW 160
#define NPTS (BATCH*DD*HH*WW)   /* 9,830,400 — divisible by 256, no tail */
#define TPB  256
#define NBLK (NPTS/TPB)         /* 38,400 blocks */

__device__ __forceinline__ v2f lerp2(v2f a, v2f b, float wa, float wb) {
  v2f r;
  r.x = fmaf(b.x, wb, a.x * wa);
  r.y = fmaf(b.y, wb, a.y * wa);
  return r;
}

__global__ __launch_bounds__(TPB) void st_warp_kernel(
    const float* __restrict__ vol,    // [2,160,192,160,2]
    const float* __restrict__ flow,   // [2,160,192,160,3]
    float* __restrict__ out)          // [2,160,192,160,2]
{
  // Block-local staging of this block's flow triples: 256 pts * 12 B = 3072 B,
  // fully contiguous in global memory (flow is dense over (b,d,h,w)).
  __shared__ __align__(16) float s_flow[TPB * 3];

  const int tid = threadIdx.x;
  const int p0  = (int)blockIdx.x * TPB;

  // CDNA5 async DMA: 192 lanes x B128 = 3072 B global -> LDS, tracked by
  // ASYNCcnt. th:TH_LOAD_NT: flow is a read-once stream — don't let it evict
  // the L2-resident vol (78.6 MB living in the 192 MB L2).
  {
    const float* fbase = flow + (size_t)p0 * 3;   // block-uniform -> SGPR pair (GVS mode)
    if (tid < 192) {                              // waves 0..5 fully active, 6..7 idle
      unsigned lds_off = (unsigned)(uintptr_t)(&s_flow[0]) + (unsigned)(tid * 16);
      unsigned voff    = (unsigned)(tid * 16);
      asm volatile("global_load_async_to_lds_b128 %0, %1, %2 th:TH_LOAD_NT"
                   :: "v"(lds_off), "v"(voff), "s"(fbase)
                   : "memory");
    }
  }

  // Index math is flow-independent: do it while the DMA is in flight.
  const int p = p0 + tid;
  const int b = p / (DD*HH*WW);
  int r       = p - b * (DD*HH*WW);
  const int d = r / (HH*WW);
  r           = r - d * (HH*WW);
  const int h = r / WW;
  const int w = r - h * WW;
  const v2f* __restrict__ vb = (const v2f*)vol + (size_t)b * (DD*HH*WW);

  asm volatile("s_wait_asynccnt 0" ::: "memory");
  __syncthreads();

  const float fd = s_flow[tid*3+0];
  const float fh = s_flow[tid*3+1];
  const float fw = s_flow[tid*3+2];

  const float ldc = (float)d + fd;
  const float lhc = (float)h + fh;
  const float lwc = (float)w + fw;

  // voxelmorph convention: w0 = clip(floor+1) - loc (corner bit 0 -> clip(floor)),
  // w1 = 1 - w0 (corner bit 1 -> clip(floor+1)); clamped indices stay in range.
  float f;
  f = floorf(ldc);
  const float d0f = fminf(fmaxf(f,       0.f), (float)(DD-1));
  const float d1f = fminf(fmaxf(f + 1.f, 0.f), (float)(DD-1));
  const float wd0 = d1f - ldc, wd1 = 1.f - wd0;
  const int   id0 = (int)d0f,  id1 = (int)d1f;

  f = floorf(lhc);
  const float h0f = fminf(fmaxf(f,       0.f), (float)(HH-1));
  const float h1f = fminf(fmaxf(f + 1.f, 0.f), (float)(HH-1));
  const float wh0 = h1f - lhc, wh1 = 1.f - wh0;
  const int   ih0 = (int)h0f,  ih1 = (int)h1f;

  f = floorf(lwc);
  const float w0f = fminf(fmaxf(f,       0.f), (float)(WW-1));
  const float w1f = fminf(fmaxf(f + 1.f, 0.f), (float)(WW-1));
  const float ww0 = w1f - lwc, ww1 = 1.f - ww0;
  const int   iw0 = (int)w0f,  iw1 = (int)w1f;

  // 8 corner gathers as float2 (channel pair) -> global_load_b64 with default
  // RT temporal policy: vol reuse (~8x per voxel) is served from L2.
  const int r00 = (id0*HH + ih0)*WW;
  const int r01 = (id0*HH + ih1)*WW;
  const int r10 = (id1*HH + ih0)*WW;
  const int r11 = (id1*HH + ih1)*WW;

  const v2f v000 = vb[r00+iw0], v001 = vb[r00+iw1];
  const v2f v010 = vb[r01+iw0], v011 = vb[r01+iw1];
  const v2f v100 = vb[r10+iw0], v101 = vb[r10+iw1];
  const v2f v110 = vb[r11+iw0], v111 = vb[r11+iw1];

  const v2f a00 = lerp2(v000, v001, ww0, ww1);
  const v2f a01 = lerp2(v010, v011, ww0, ww1);
  const v2f a10 = lerp2(v100, v101, ww0, ww1);
  const v2f a11 = lerp2(v110, v111, ww0, ww1);
  const v2f c0  = lerp2(a00, a01, wh0, wh1);
  const v2f c1  = lerp2(a10, a11, wh0, wh1);
  const v2f res = lerp2(c0, c1, wd0, wd1);

  // Write-once stream: non-temporal so vol keeps the L2.
  __builtin_nontemporal_store(res, (v2f*)out + p);
}

extern "C" void kernel_launch(void* const* d_in, const int* in_sizes, int n_in,
                              void* d_out, int out_size, void* d_ws, size_t ws_size,
                              hipStream_t stream) {
  (void)in_sizes; (void)n_in; (void)out_size; (void)d_ws; (void)ws_size;
  const float* vol  = (const float*)d_in[0];
  const float* flow = (const float*)d_in[1];
  float*       out  = (float*)d_out;
  st_warp_kernel<<<NBLK, TPB, 0, stream>>>(vol, flow, out);
}